// ReducedTensorProducts_56899726737858
// MI455X (gfx1250) — compile-verified
//
#include <hip/hip_runtime.h>

typedef __attribute__((ext_vector_type(2))) float v2f;
typedef __attribute__((ext_vector_type(8))) float v8f;

#define BATCH   256
#define DDIM    128
#define ZDIM    8256
#define NCT     16                 // 16 column tiles of 16 batches = 256
#define NWAVES  4
#define IPW     (DDIM / NWAVES)    // 32 i-values per wave (K-split)
#define CTSTR   132                // per-ct stride (floats): bank step 4
#define SLO     (NCT * CTSTR + 4)  // 2116 floats per lo-row: bank step 4 per lane

extern __shared__ float lds[];

__global__ void __launch_bounds__(128, 1)
rtp_wmma_f32(const float* __restrict__ x,
             const float* __restrict__ y,
             const float* __restrict__ C,
             float* __restrict__ out)
{
    const int tid  = threadIdx.x;
    const int lane = tid & 31;
    const int wave = tid >> 5;
    const int lo   = lane & 15;    // A row / B col within the 16x16 tile
    const int hi   = lane >> 4;    // selects K pair {0,1} vs {2,3}
    const int z0   = blockIdx.x << 4;

    // ---- stage y into LDS: addr(b,j) = (b&15)*SLO + (b>>4)*CTSTR + j ----
    // Puts all 16 ct-slices a lane needs within DS immediate-offset range.
    float* ys = lds;
    for (int idx = tid; idx < BATCH * DDIM; idx += 128) {
        const int b = idx >> 7;
        const int j = idx & (DDIM - 1);
        ys[(b & 15) * SLO + (b >> 4) * CTSTR + j] = y[idx];
    }
    __syncthreads();

    v8f acc[NCT] = {};

    const float* yb = ys + lo * SLO + (hi << 1);                       // one DS base
    const float* cb = C + (size_t)(z0 + lo) * (DDIM * DDIM) + (hi << 1); // A-row base
    const float* xb = x + lo * DDIM;                                   // x base
    const int i0 = wave * IPW;

    for (int i = i0; i < i0 + IPW; ++i) {
        // hoist x[b][i] for all 16 column tiles: one base + imm offsets (ct*8192B)
        float xv[NCT];
#pragma unroll
        for (int ct = 0; ct < NCT; ++ct)
            xv[ct] = xb[ct * 16 * DDIM + i];

        const float* cptr = cb + i * DDIM;
        __builtin_prefetch(cptr + DDIM, 0, 1);   // next C row -> global_prefetch_b8

#pragma unroll 8
        for (int j = 0; j < DDIM; j += 4) {
            // A fragment: 8 bytes of C per lane, reused by 16 WMMAs
            const v2f a = *(const v2f*)(cptr + j);
#pragma unroll
            for (int ct = 0; ct < NCT; ++ct) {
                // B fragment on the fly: P[b][k] = x[b][i] * y[b][j+k]
                const v2f yv = *(const v2f*)(yb + ct * CTSTR + j);
                v2f bf;
                bf.x = xv[ct] * yv.x;
                bf.y = xv[ct] * yv.y;
                acc[ct] = __builtin_amdgcn_wmma_f32_16x16x4_f32(
                    false, a, false, bf, (short)0, acc[ct], false, false);
            }
        }
    }

    __syncthreads();

    // ---- cross-wave K reduction through LDS (reuse staging space) ----
    float* red = lds;   // layout: [wave][ct][m:16][n:16]
#pragma unroll
    for (int ct = 0; ct < NCT; ++ct) {
#pragma unroll
        for (int v = 0; v < 8; ++v) {
            const int m = v + (hi << 3);
            red[((wave * NCT + ct) * 16 + m) * 16 + lo] = acc[ct][v];
        }
    }
    __syncthreads();

    for (int idx = tid; idx < NCT * 256; idx += 128) {
        const int ct = idx >> 8;
        const int r  = idx & 255;
        const int m  = r >> 4;
        const int n  = r & 15;
        float s = 0.f;
#pragma unroll
        for (int w = 0; w < NWAVES; ++w)
            s += red[((w * NCT + ct) * 16 + m) * 16 + n];
        out[(size_t)(ct * 16 + n) * ZDIM + (z0 + m)] = s;
    }
}

extern "C" void kernel_launch(void* const* d_in, const int* in_sizes, int n_in,
                              void* d_out, int out_size, void* d_ws, size_t ws_size,
                              hipStream_t stream)
{
    const float* x = (const float*)d_in[0];
    const float* y = (const float*)d_in[1];
    const float* C = (const float*)d_in[2];
    float* out = (float*)d_out;
    const size_t shmem = (size_t)NCT * SLO * sizeof(float); // 135424 B (< 320 KB WGP LDS)
    rtp_wmma_f32<<<dim3(ZDIM / 16), dim3(128), shmem, stream>>>(x, y, C, out);
}